// ConvGCN_3822520893972
// MI455X (gfx1250) — compile-verified
//
#include <hip/hip_runtime.h>
#include <hip/hip_bf16.h>

typedef float v2f __attribute__((ext_vector_type(2)));
typedef float v8f __attribute__((ext_vector_type(8)));

#define N_NODES 50000
#define N_EDGES 800000

// ---------------------------------------------------------------------------
// Degree / D^-1/2 kernels (self-loop counted by initializing deg to 1)
// ---------------------------------------------------------------------------
__global__ void k_init_deg(float* __restrict__ deg, int n) {
  int i = blockIdx.x * blockDim.x + threadIdx.x;
  if (i < n) deg[i] = 1.0f;  // self loop
}

__global__ void k_count_deg(const int* __restrict__ dst, float* __restrict__ deg, int e) {
  int i = blockIdx.x * blockDim.x + threadIdx.x;
  if (i < e) atomicAdd(&deg[dst[i]], 1.0f);
}

__global__ void k_dinv(float* __restrict__ deg, int n) {
  int i = blockIdx.x * blockDim.x + threadIdx.x;
  if (i < n) deg[i] = rsqrtf(deg[i]);  // deg >= 1 always
}

// ---------------------------------------------------------------------------
// Repack W [K, N] row-major into WMMA B-fragment order:
//   Wp[k>>2][n][k&3]  (so each lane's B frag is one contiguous b64 load)
// ---------------------------------------------------------------------------
__global__ void k_pack_w(const float* __restrict__ W, float* __restrict__ Wp,
                         int K, int N) {
  int idx = blockIdx.x * blockDim.x + threadIdx.x;
  if (idx >= K * N) return;
  int k = idx / N;
  int n = idx - k * N;
  Wp[(size_t)(k >> 2) * (N * 4) + n * 4 + (k & 3)] = W[idx];
}

// ---------------------------------------------------------------------------
// WMMA f32 GEMM: out[r, out_off + c] = A[r,:] @ W[:,c]  (+bias, ReLU optional)
// A: [M, KDIM] row-major, Wp: packed fragment order (see k_pack_w).
// Block = 128 threads = 4 waves; each wave computes a 16 x (NT*16) tile using
// V_WMMA_F32_16X16X4_F32, K stepped by 4 (unrolled 4x).
// A frag (16x4 f32): lanes 0-15 hold K={k0,k0+1}, lanes 16-31 hold K={k0+2,k0+3}
// B frag (4x16 f32): vgpr j, lane l -> K = k0 + (l>>4)*2 + j, N = l&15
// C/D (16x16 f32): vgpr i, lanes 0-15 -> M=i, lanes 16-31 -> M=i+8, N = l&15
// ---------------------------------------------------------------------------
template <int KDIM, int NT, bool BIAS_RELU>
__global__ __launch_bounds__(128) void k_gemm_wmma(
    const float* __restrict__ A, const float* __restrict__ Wp,
    const float* __restrict__ bias, float* __restrict__ out,
    int M, int out_stride, int out_off) {
  const int lane = threadIdx.x & 31;
  const int wave = threadIdx.x >> 5;
  const int row_tile = (blockIdx.x * 4 + wave) * 16;
  if (row_tile >= M) return;  // wave-uniform: EXEC stays all-1s for WMMA

  const int m16 = lane & 15;
  const int khalf = lane >> 4;  // 0 -> K pair {0,1}; 1 -> {2,3}
  constexpr int NCOLS = NT * 16;

  v8f acc[NT];
#pragma unroll
  for (int t = 0; t < NT; ++t) acc[t] = (v8f){0.f, 0.f, 0.f, 0.f, 0.f, 0.f, 0.f, 0.f};

  const float* __restrict__ arow = A + (size_t)(row_tile + m16) * KDIM + khalf * 2;
  const float* __restrict__ wlane = Wp + m16 * 4 + khalf * 2;

  for (int k0 = 0; k0 < KDIM; k0 += 16) {
    // Sequential A-row stream: speculative prefetch 1KB ahead (global_prefetch_b8)
    if (k0 + 256 < KDIM) __builtin_prefetch(arow + k0 + 256, 0, 0);
#pragma unroll
    for (int kq = 0; kq < 4; ++kq) {
      const int k = k0 + kq * 4;
      v2f afrag = *(const v2f*)(arow + k);                       // b64, 8B aligned
      const float* wq = wlane + (size_t)(k >> 2) * (NCOLS * 4);
#pragma unroll
      for (int t = 0; t < NT; ++t) {
        v2f bfrag = *(const v2f*)(wq + t * 64);                  // b64, 8B aligned
        acc[t] = __builtin_amdgcn_wmma_f32_16x16x4_f32(
            /*neg_a=*/false, afrag, /*neg_b=*/false, bfrag,
            /*c_mod=*/(short)0, acc[t], /*reuse_a=*/false, /*reuse_b=*/false);
      }
    }
  }

  const int rbase = row_tile + khalf * 8;
#pragma unroll
  for (int t = 0; t < NT; ++t) {
#pragma unroll
    for (int i = 0; i < 8; ++i) {
      const int c = t * 16 + m16;
      float v = acc[t][i];
      if (BIAS_RELU) v = fmaxf(v + bias[c], 0.0f);
      out[(size_t)(rbase + i) * out_stride + out_off + c] = v;
    }
  }
}

// ---------------------------------------------------------------------------
// Self-loop + bias init: out[i,c] = b[c] + xw[i,c] * dinv[i]^2
// ---------------------------------------------------------------------------
__global__ void k_node_init(const float* __restrict__ xw, const float* __restrict__ dinv,
                            const float* __restrict__ bias, float* __restrict__ out,
                            int n, int C) {
  int idx = blockIdx.x * blockDim.x + threadIdx.x;
  if (idx >= n * C) return;
  int i = idx / C;
  int c = idx - i * C;
  float di = dinv[i];
  out[idx] = bias[c] + xw[idx] * di * di;
}

// ---------------------------------------------------------------------------
// Edge scatter: out[dst] += xw[src] * dinv[src]*dinv[dst]
// One thread per (edge, 4-col chunk): float4 gather (L2-resident) + 4 atomics.
// ---------------------------------------------------------------------------
template <int C>
__global__ void k_scatter(const float* __restrict__ xw, const float* __restrict__ dinv,
                          const int* __restrict__ src, const int* __restrict__ dst,
                          float* __restrict__ out, int e) {
  constexpr int CH = C / 4;
  int tid = blockIdx.x * blockDim.x + threadIdx.x;
  int eidx = tid / CH;
  int ch = tid - eidx * CH;
  if (eidx >= e) return;
  int s = src[eidx];
  int d = dst[eidx];
  float nrm = dinv[s] * dinv[d];
  float4 v = *((const float4*)(xw + (size_t)s * C) + ch);
  float* o = out + (size_t)d * C + ch * 4;
  atomicAdd(o + 0, v.x * nrm);
  atomicAdd(o + 1, v.y * nrm);
  atomicAdd(o + 2, v.z * nrm);
  atomicAdd(o + 3, v.w * nrm);
}

__global__ void k_relu(float* __restrict__ x, int n) {
  int i = blockIdx.x * blockDim.x + threadIdx.x;
  if (i < n) x[i] = fmaxf(x[i], 0.0f);
}

// ---------------------------------------------------------------------------
extern "C" void kernel_launch(void* const* d_in, const int* in_sizes, int n_in,
                              void* d_out, int out_size, void* d_ws, size_t ws_size,
                              hipStream_t stream) {
  (void)in_sizes; (void)n_in; (void)out_size; (void)ws_size;

  const float* feature = (const float*)d_in[0];   // [N, 512]
  const float* img     = (const float*)d_in[1];   // [N, 256]
  const int*   edges   = (const int*)d_in[2];     // [2, E]
  const float* W_feat  = (const float*)d_in[3];   // [512, 64]
  const float* b_feat  = (const float*)d_in[4];
  const float* W_img   = (const float*)d_in[5];   // [256, 32]
  const float* b_img   = (const float*)d_in[6];
  const float* W_g1    = (const float*)d_in[7];   // [96, 96]
  const float* b_g1    = (const float*)d_in[8];
  const float* W_g2    = (const float*)d_in[9];   // [96, 64]
  const float* b_g2    = (const float*)d_in[10];
  float* out = (float*)d_out;                     // [N, 64]

  const int* src = edges;
  const int* dst = edges + N_EDGES;

  // Workspace layout (floats):
  //   dinv[50048] | bufA[N*96] | bufB[N*96] | packed weights (~56K floats)
  float* ws     = (float*)d_ws;
  float* dinv   = ws;
  float* bufA   = ws + 50048;                        // x, later reused for h
  float* bufB   = bufA + (size_t)N_NODES * 96;       // xw1, later xw2
  float* wpF    = bufB + (size_t)N_NODES * 96;       // 512*64
  float* wpI    = wpF + 512 * 64;                    // 256*32
  float* wpG1   = wpI + 256 * 32;                    // 96*96
  float* wpG2   = wpG1 + 96 * 96;                    // 96*64

  const int T = 256;
  const int MB = (N_NODES + 63) / 64;  // 782 blocks * 4 waves * 16 rows

  // Pack all weights into WMMA fragment order (tiny one-time passes)
  k_pack_w<<<(512 * 64 + T - 1) / T, T, 0, stream>>>(W_feat, wpF, 512, 64);
  k_pack_w<<<(256 * 32 + T - 1) / T, T, 0, stream>>>(W_img, wpI, 256, 32);
  k_pack_w<<<(96 * 96 + T - 1) / T, T, 0, stream>>>(W_g1, wpG1, 96, 96);
  k_pack_w<<<(96 * 64 + T - 1) / T, T, 0, stream>>>(W_g2, wpG2, 96, 64);

  // D^-1/2 with self loops
  k_init_deg<<<(N_NODES + T - 1) / T, T, 0, stream>>>(dinv, N_NODES);
  k_count_deg<<<(N_EDGES + T - 1) / T, T, 0, stream>>>(dst, dinv, N_EDGES);
  k_dinv<<<(N_NODES + T - 1) / T, T, 0, stream>>>(dinv, N_NODES);

  // Encoders -> x = bufA [N, 96] (cols 0-63 feat, 64-95 img), bias+ReLU fused
  k_gemm_wmma<512, 4, true><<<MB, 128, 0, stream>>>(feature, wpF, b_feat, bufA, N_NODES, 96, 0);
  k_gemm_wmma<256, 2, true><<<MB, 128, 0, stream>>>(img, wpI, b_img, bufA, N_NODES, 96, 64);

  // Layer 1: xw1 = x @ W_g1 -> bufB [N, 96]
  k_gemm_wmma<96, 6, false><<<MB, 128, 0, stream>>>(bufA, wpG1, nullptr, bufB, N_NODES, 96, 0);
  // h (bufA) = b_g1 + self-loop + edge aggregation, then ReLU
  k_node_init<<<(N_NODES * 96 + T - 1) / T, T, 0, stream>>>(bufB, dinv, b_g1, bufA, N_NODES, 96);
  k_scatter<96><<<(N_EDGES * 24 + T - 1) / T, T, 0, stream>>>(bufB, dinv, src, dst, bufA, N_EDGES);
  k_relu<<<(N_NODES * 96 + T - 1) / T, T, 0, stream>>>(bufA, N_NODES * 96);

  // Layer 2: xw2 = h @ W_g2 -> bufB [N, 64]
  k_gemm_wmma<96, 4, false><<<MB, 128, 0, stream>>>(bufA, wpG2, nullptr, bufB, N_NODES, 64, 0);
  k_node_init<<<(N_NODES * 64 + T - 1) / T, T, 0, stream>>>(bufB, dinv, b_g2, out, N_NODES, 64);
  k_scatter<64><<<(N_EDGES * 16 + T - 1) / T, T, 0, stream>>>(bufB, dinv, src, dst, out, N_EDGES);
}